// BiGramWithSelfAttentionLanguageModel_48292612276278
// MI455X (gfx1250) — compile-verified
//
#include <hip/hip_runtime.h>
#include <math.h>

// ---------------------------------------------------------------------------
// CDNA5 (gfx1250) wave32 WMMA f32 path:  D = A(16x4) * B(4x16) + C(16x16)
// A frag: v2f  lane<16 -> A[M=lane, K=0..1], lane>=16 -> A[M=lane-16, K=2..3]
// B frag: v2f  lane<16 -> B[K=0..1, N=lane], lane>=16 -> B[K=2..3, N=lane-16]
// C/D   : v8f  vgpr r  -> M = r + 8*(lane>=16), N = lane&15
// ---------------------------------------------------------------------------
typedef __attribute__((ext_vector_type(2))) float v2f;
typedef __attribute__((ext_vector_type(8))) float v8f;
typedef __attribute__((ext_vector_type(4))) int   v4i;

__device__ __forceinline__ v8f wmma_f32(v2f a, v2f b, v8f c) {
  return __builtin_amdgcn_wmma_f32_16x16x4_f32(
      /*neg_a=*/false, a, /*neg_b=*/false, b,
      /*c_mod=*/(short)0, c, /*reuse_a=*/false, /*reuse_b=*/false);
}

// ds_swizzle_b32 XOR-mode lane exchange (no index VGPRs, immediate-encoded).
// group-of-32 encoding: offset[14:10]=xor_mask, offset[9:5]=or, offset[4:0]=and
template <int XM>
__device__ __forceinline__ float swz_xor(float v) {
  int i = __builtin_bit_cast(int, v);
  i = __builtin_amdgcn_ds_swizzle(i, (XM << 10) | 0x1F);
  return __builtin_bit_cast(float, i);
}

// --- gfx1250 async global->LDS copy (ASYNCcnt) with safe fallback ----------
#if __has_builtin(__builtin_amdgcn_global_load_async_to_lds_b128)
#define HAVE_ASYNC_LDS 1
__device__ __forceinline__ void async_copy16(void* lds_dst, const void* gsrc) {
  __builtin_amdgcn_global_load_async_to_lds_b128(
      (v4i*)gsrc, (v4i*)lds_dst, 0, 0);
}
__device__ __forceinline__ void async_wait0() {
#if __has_builtin(__builtin_amdgcn_s_wait_asynccnt)
  __builtin_amdgcn_s_wait_asynccnt(0);
#else
  asm volatile("s_wait_asynccnt 0" ::: "memory");
#endif
}
#else
#define HAVE_ASYNC_LDS 0
#endif

#define N_EMBED 256
#define VOCAB   32000
#define SEQ_T   1024
#define BATCH   4
#define ROWS    (BATCH * SEQ_T)   // 4096

// ---------------------------------------------------------------------------
// 1) x[row, :] = tok_table[idx[row], :] + pos_table[row % T, :]
// ---------------------------------------------------------------------------
__global__ void embed_kernel(const int* __restrict__ idx,
                             const float* __restrict__ tok,
                             const float* __restrict__ pos,
                             float* __restrict__ x) {
  const int row = blockIdx.x;            // 0..4095
  const int t   = row & (SEQ_T - 1);
  const int tk  = idx[row];
  const int c   = threadIdx.x;           // 0..255
  x[(size_t)row * N_EMBED + c] =
      tok[(size_t)tk * N_EMBED + c] + pos[(size_t)t * N_EMBED + c];
}

// ---------------------------------------------------------------------------
// 2) f32 WMMA GEMM (N compile-time):  C[M,N] = A[M,256] * B[256,N] (+ bias)
//    Block = 256 threads = 8 waves; one 16-row M tile, 8 x 16-col N tiles.
//    A tile staged in LDS via async global->LDS (padded stride 260).
//    grid = (N/16/8, M/16)
// ---------------------------------------------------------------------------
template <int N>
__global__ void gemm_wmma_kernel(const float* __restrict__ A,
                                 const float* __restrict__ Bm,
                                 const float* __restrict__ bias,
                                 float* __restrict__ C) {
  __shared__ float As[16][260];          // 16x256 tile, padded vs bank conflicts

  const int tileM = blockIdx.y << 4;
  const int wave  = threadIdx.x >> 5;
  const int lane  = threadIdx.x & 31;
  const int half  = lane >> 4;
  const int l16   = lane & 15;
  const int tileN = ((blockIdx.x << 3) + wave) << 4;
  const int kb    = half << 1;           // per-lane K sub-offset (0 or 2)

  // stage A tile [16][256]: 4096 floats, 4 x 16B per thread
  {
    const int t = threadIdx.x;
#pragma unroll
    for (int j = 0; j < 4; ++j) {
      const int l = (t << 2) + (j << 10);      // element index in [0,4096)
      const int m = l >> 8;
      const int c = l & 255;
      const float* gp = &A[(size_t)(tileM + m) * N_EMBED + c];
#if HAVE_ASYNC_LDS
      async_copy16(&As[m][c], gp);             // global_load_async_to_lds_b128
#else
      *reinterpret_cast<float4*>(&As[m][c]) =
          *reinterpret_cast<const float4*>(gp);
#endif
    }
  }
#if HAVE_ASYNC_LDS
  async_wait0();
#endif
  __syncthreads();

  v8f acc = {};
  const float* bp = Bm + (size_t)kb * N + tileN + l16;   // per-lane B base

#pragma unroll 8
  for (int k = 0; k < N_EMBED; k += 4) {
    v2f a, b;
    a.x = As[l16][k + kb];
    a.y = As[l16][k + kb + 1];
    b.x = bp[(size_t)k * N];
    b.y = bp[(size_t)k * N + N];
    acc = wmma_f32(a, b, acc);
  }

  const float bb = bias ? bias[tileN + l16] : 0.0f;
  float* cp = C + (size_t)(tileM + (half << 3)) * N + tileN + l16;
#pragma unroll
  for (int r = 0; r < 8; ++r) {
    cp[(size_t)r * N] = acc[r] + bb;
  }
}

// ---------------------------------------------------------------------------
// 3) Causal flash attention, one wave32 per (batch, 16-query tile).
//    Q tile staged in LDS once (async), S = Q K^T * 1/16 with A-frags from
//    LDS, online softmax (ds_swizzle XOR reductions), O += P V.
//    O tile = 16 v8f accumulators.  grid = (64, 4), block = 32.
// ---------------------------------------------------------------------------
__global__ void __launch_bounds__(32, 1)
attn_kernel(const float* __restrict__ Q,
            const float* __restrict__ K,
            const float* __restrict__ V,
            float* __restrict__ O) {
  __shared__ float Qs[16][260];          // 16x256 Q tile, padded
  __shared__ float Ps[16][17];           // P transpose staging

  const int qt   = blockIdx.x;           // query tile within batch, 0..63
  const int b    = blockIdx.y;           // batch 0..3
  const int lane = threadIdx.x & 31;
  const int half = lane >> 4;
  const int l16  = lane & 15;
  const int kb   = half << 1;
  const int rowBase = b * SEQ_T;         // global row base for this batch
  const int qRow0   = qt << 4;

  // ---- stage Q tile [16][256] into LDS: 32 lanes x 32 x 16B ----
  {
    const int t = threadIdx.x;           // 0..31
#pragma unroll
    for (int j = 0; j < 32; ++j) {
      const int l = (t << 2) + (j << 7); // element index in [0,4096)
      const int m = l >> 8;
      const int c = l & 255;
      const float* gp = &Q[(size_t)(rowBase + qRow0 + m) * N_EMBED + c];
#if HAVE_ASYNC_LDS
      async_copy16(&Qs[m][c], gp);
#else
      *reinterpret_cast<float4*>(&Qs[m][c]) =
          *reinterpret_cast<const float4*>(gp);
#endif
    }
  }
#if HAVE_ASYNC_LDS
  async_wait0();
#endif
  __syncthreads();

  v8f accO[16];
#pragma unroll
  for (int i = 0; i < 16; ++i) accO[i] = (v8f){};
  float mrow[8], lrow[8];
#pragma unroll
  for (int r = 0; r < 8; ++r) { mrow[r] = -3.0e38f; lrow[r] = 0.0f; }

  for (int kt = 0; kt <= qt; ++kt) {
    const float* kp = K + (size_t)(rowBase + (kt << 4) + l16) * N_EMBED + kb;

    // ---- S tile = Q(16x256) * K^T(256x16), A-frags from LDS ----
    v8f s = {};
#pragma unroll 8
    for (int k = 0; k < N_EMBED; k += 4) {
      v2f a, bk;
      a.x = Qs[l16][k + kb];
      a.y = Qs[l16][k + kb + 1];
      bk.x = kp[k];
      bk.y = kp[k + 1];
      s = wmma_f32(a, bk, s);
    }

    // ---- scale, causal mask, online softmax stats + O rescale ----
#pragma unroll
    for (int r = 0; r < 8; ++r) {
      const int qrow = qRow0 + r + (half << 3);
      const int kcol = (kt << 4) + l16;
      float sv = s[r] * 0.0625f;               // 1/sqrt(256)
      if (kcol > qrow) sv = -3.0e38f;
      float mx = sv;                            // max over 16 cols (lanes)
      mx = fmaxf(mx, swz_xor<1>(mx));
      mx = fmaxf(mx, swz_xor<2>(mx));
      mx = fmaxf(mx, swz_xor<4>(mx));
      mx = fmaxf(mx, swz_xor<8>(mx));
      const float mnew = fmaxf(mrow[r], mx);
      const float al   = __expf(mrow[r] - mnew);
      const float p    = __expf(sv - mnew);
      float ps = p;                             // sum over 16 cols
      ps += swz_xor<1>(ps);
      ps += swz_xor<2>(ps);
      ps += swz_xor<4>(ps);
      ps += swz_xor<8>(ps);
      lrow[r] = lrow[r] * al + ps;
      mrow[r] = mnew;
#pragma unroll
      for (int nt = 0; nt < 16; ++nt) accO[nt][r] *= al;
      Ps[r + (half << 3)][l16] = p;             // stash P in C-layout
    }
    __syncthreads();   // single-wave WG: S_NOP + compiler LDS fence

    // ---- re-read P as A fragments (transposed layout) ----
    v2f ap[4];
#pragma unroll
    for (int ks = 0; ks < 4; ++ks) {
      ap[ks].x = Ps[l16][(ks << 2) + kb];
      ap[ks].y = Ps[l16][(ks << 2) + kb + 1];
    }
    __syncthreads();

    // ---- O += P(16x16) * V(16x256) ----
    const float* vp0 =
        V + (size_t)(rowBase + (kt << 4) + kb) * N_EMBED + l16;
#pragma unroll
    for (int nt = 0; nt < 16; ++nt) {
#pragma unroll
      for (int ks = 0; ks < 4; ++ks) {
        v2f bv;
        const float* vp = vp0 + (size_t)(ks << 2) * N_EMBED + (nt << 4);
        bv.x = vp[0];
        bv.y = vp[N_EMBED];
        accO[nt] = wmma_f32(ap[ks], bv, accO[nt]);
      }
    }
  }

  // ---- normalize and store ----
  float* op = O + (size_t)(rowBase + qRow0 + (half << 3)) * N_EMBED + l16;
#pragma unroll
  for (int nt = 0; nt < 16; ++nt) {
#pragma unroll
    for (int r = 0; r < 8; ++r) {
      op[(size_t)r * N_EMBED + (nt << 4)] = accO[nt][r] / lrow[r];
    }
  }
}

// ---------------------------------------------------------------------------
// 4) per-row loss: rowloss[row] = logsumexp(logits[row,:]) - logits[row,tgt]
// ---------------------------------------------------------------------------
__global__ void row_loss_kernel(const float* __restrict__ logits,
                                const int* __restrict__ tgt,
                                float* __restrict__ rowloss) {
  __shared__ float sm[256], ss[256];
  const int row = blockIdx.x;
  const float* lr = logits + (size_t)row * VOCAB;

  float m = -3.0e38f, s = 0.0f;
  for (int c = threadIdx.x; c < VOCAB; c += 256) {
    const float xv = lr[c];
    if (xv > m) {
      s = s * __expf(m - xv) + 1.0f;
      m = xv;
    } else {
      s += __expf(xv - m);
    }
  }
  sm[threadIdx.x] = m;
  ss[threadIdx.x] = s;
  __syncthreads();
  for (int off = 128; off > 0; off >>= 1) {
    if (threadIdx.x < off) {
      const float m1 = sm[threadIdx.x],       s1 = ss[threadIdx.x];
      const float m2 = sm[threadIdx.x + off], s2 = ss[threadIdx.x + off];
      const float mn = fmaxf(m1, m2);
      sm[threadIdx.x] = mn;
      ss[threadIdx.x] = s1 * __expf(m1 - mn) + s2 * __expf(m2 - mn);
    }
    __syncthreads();
  }
  if (threadIdx.x == 0) {
    const float lse = sm[0] + __logf(ss[0]);
    rowloss[row] = lse - lr[tgt[row]];
  }
}

// ---------------------------------------------------------------------------
// 5) loss = mean(rowloss)
// ---------------------------------------------------------------------------
__global__ void loss_reduce_kernel(const float* __restrict__ rowloss,
                                   float* __restrict__ out) {
  __shared__ float sh[256];
  float s = 0.0f;
  for (int i = threadIdx.x; i < ROWS; i += 256) s += rowloss[i];
  sh[threadIdx.x] = s;
  __syncthreads();
  for (int off = 128; off > 0; off >>= 1) {
    if (threadIdx.x < off) sh[threadIdx.x] += sh[threadIdx.x + off];
    __syncthreads();
  }
  if (threadIdx.x == 0) out[0] = sh[0] * (1.0f / (float)ROWS);
}

// ---------------------------------------------------------------------------
extern "C" void kernel_launch(void* const* d_in, const int* in_sizes, int n_in,
                              void* d_out, int out_size, void* d_ws, size_t ws_size,
                              hipStream_t stream) {
  (void)in_sizes; (void)n_in; (void)out_size; (void)ws_size;

  const int*   idx = (const int*)d_in[0];
  const int*   tgt = (const int*)d_in[1];
  const float* tok = (const float*)d_in[2];
  const float* pos = (const float*)d_in[3];
  const float* Wk  = (const float*)d_in[4];
  const float* Wq  = (const float*)d_in[5];
  const float* Wv  = (const float*)d_in[6];
  const float* lmW = (const float*)d_in[7];
  const float* lmb = (const float*)d_in[8];

  float* logits = (float*)d_out;                          // [4096, 32000]
  float* loss   = logits + (size_t)ROWS * VOCAB;          // final scalar

  float* x     = (float*)d_ws;                            // [4096,256]
  float* q     = x    + (size_t)ROWS * N_EMBED;
  float* k     = q    + (size_t)ROWS * N_EMBED;
  float* v     = k    + (size_t)ROWS * N_EMBED;
  float* attn  = v    + (size_t)ROWS * N_EMBED;
  float* rloss = attn + (size_t)ROWS * N_EMBED;           // [4096]

  // 1) embeddings
  embed_kernel<<<ROWS, 256, 0, stream>>>(idx, tok, pos, x);

  // 2) Q, K, V projections: M=4096, N=256 -> grid (16 Ntiles / 8, 256 Mtiles)
  dim3 gqkv(N_EMBED / 16 / 8, ROWS / 16);
  gemm_wmma_kernel<N_EMBED><<<gqkv, 256, 0, stream>>>(x, Wq, nullptr, q);
  gemm_wmma_kernel<N_EMBED><<<gqkv, 256, 0, stream>>>(x, Wk, nullptr, k);
  gemm_wmma_kernel<N_EMBED><<<gqkv, 256, 0, stream>>>(x, Wv, nullptr, v);

  // 3) causal flash attention
  attn_kernel<<<dim3(SEQ_T / 16, BATCH), 32, 0, stream>>>(q, k, v, attn);

  // 4) logits GEMM + bias: M=4096, N=32000 -> grid (2000/8, 256)
  dim3 glog(VOCAB / 16 / 8, ROWS / 16);
  gemm_wmma_kernel<VOCAB><<<glog, 256, 0, stream>>>(attn, lmW, lmb, logits);

  // 5) cross-entropy
  row_loss_kernel<<<ROWS, 256, 0, stream>>>(logits, tgt, rloss);
  loss_reduce_kernel<<<1, 256, 0, stream>>>(rloss, loss);
}